// GNet_6691559047485
// MI455X (gfx1250) — compile-verified
//
#include <hip/hip_runtime.h>
#include <hip/hip_bf16.h>

// ---------------------------------------------------------------------------
// MetaLayer GNN (3 layers) for MI455X / gfx1250.
// All MLPs run on V_WMMA_F32_16X16X4_F32 (exact fp32 => matches reference
// numerics; workload is gather-bound so low-precision WMMA buys nothing).
// One wave32 processes 16 rows (edges/nodes/batches) per MLP chain.
// Weights are staged once per block into LDS, zero-padded and K-pair
// interleaved, so the WMMA inner loop is ds_load_b64 + ds_load_b64 + v_wmma
// with no divergent guards.
// ---------------------------------------------------------------------------

typedef float v2f __attribute__((ext_vector_type(2)));
typedef float v8f __attribute__((ext_vector_type(8)));

#if defined(__gfx1250__) && __has_builtin(__builtin_amdgcn_wmma_f32_16x16x4_f32)
#define USE_WMMA 1
#else
#define USE_WMMA 0
#endif

constexpr int pad4(int v) { return (v + 3) & ~3; }

#define AGS 48    // node accumulator stride: [0..NO) h-sum, [32..42) edge_u-sum, [44] count
#define XS  24    // node feature buffer stride (max 23 features)
#define US  32    // u buffer stride
#define ES  10    // edge feature buffer stride
#define CC  5000  // nodes per batch
#define NN  160000L
#define EE  2560000L

// Stage W (krows x ncols, row-major global) into LDS as zero-padded,
// K-pair-interleaved layout: sw[(k/2)*NW*2 + n*2 + (k&1)] = W[k][n].
// This makes each B fragment (rows kb, kb+1 at column n) one aligned b64 load.
template <int KP, int NW>
__device__ __forceinline__ void stage_w(const float* __restrict__ Wg, int krows,
                                        int ncols, const float* __restrict__ bg,
                                        float* sw, float* sb) {
  const int tid = threadIdx.x, nth = blockDim.x;
  for (int t = tid; t < (KP / 2) * NW; t += nth) {
    const int kk = t / NW, n = t - kk * NW;
    const int k0 = 2 * kk;
    sw[t * 2 + 0] = (k0 < krows && n < ncols) ? Wg[(long)k0 * ncols + n] : 0.0f;
    sw[t * 2 + 1] = (k0 + 1 < krows && n < ncols) ? Wg[(long)(k0 + 1) * ncols + n] : 0.0f;
  }
  for (int t = tid; t < NW; t += nth) sb[t] = (t < ncols) ? bg[t] : 0.0f;
}

// 16xNCOLS = A(16xKP in LDS, zero-padded) * W(staged in LDS) + bias,
// optional ReLU. Output rows to LDS `so` (stride SO); columns [NCOLS, ZFILL)
// come out as exact zeros (padded weights/bias) so the result can feed the
// next GEMM's A directly.
template <int KP, int NCOLS, int NT, bool RELU, int ZFILL>
__device__ __forceinline__ void wave_mlp_gemm(const float* sa, int SA,
                                              const float* sw, const float* sb,
                                              float* so, int SO) {
  constexpr int NW = 16 * NT;
  const int lane = threadIdx.x & 31;
  const int half = lane >> 4;
  const int m = lane & 15;
#pragma unroll
  for (int nt = 0; nt < NT; ++nt) {
    const int n = m + nt * 16;
    float accv[8];
#if USE_WMMA
    v8f acc = {};
#pragma unroll
    for (int k0 = 0; k0 < KP; k0 += 4) {
      const int kb = k0 + 2 * half;
      // A 16x4 fp32 fragment: lanes 0-15 hold K=k0,k0+1; lanes 16-31 K=k0+2,k0+3
      v2f a = *(const v2f*)(sa + m * SA + kb);
      // B 4x16 fp32 fragment: rows kb,kb+1 at column n (pair-interleaved LDS)
      v2f b = *(const v2f*)(sw + ((k0 >> 1) + half) * NW * 2 + n * 2);
      acc = __builtin_amdgcn_wmma_f32_16x16x4_f32(false, a, false, b, (short)0,
                                                  acc, false, false);
    }
#pragma unroll
    for (int j = 0; j < 8; ++j) accv[j] = acc[j];
#else
#pragma unroll
    for (int j = 0; j < 8; ++j) {
      const int r = j + 8 * half;
      float v = 0.0f;
      for (int k = 0; k < KP; ++k)
        v += sa[r * SA + k] * sw[(k >> 1) * NW * 2 + n * 2 + (k & 1)];
      accv[j] = v;
    }
#endif
    const float bv = sb[n];
#pragma unroll
    for (int j = 0; j < 8; ++j) {
      const int r = j + 8 * half;  // D layout: VGPR j holds row j + 8*half, col n
      float v = accv[j] + bv;
      if (RELU) v = fmaxf(v, 0.0f);
      if (n < ZFILL) so[r * SO + n] = v;  // padded cols are already exact 0
    }
  }
}

// ---------------------------------------------------------------------------
// Fused edge-MLP + node1-MLP + scatter-add, 16 edges per wave, 4 waves/block.
// ---------------------------------------------------------------------------
template <int FX, int XLD, int FE, int FU, int EH, int EO, int NH, int NO, bool STORE_E>
__global__ __launch_bounds__(128) void edge_node1_kernel(
    const float* __restrict__ xin, const float* __restrict__ ein,
    float* __restrict__ eout, const float* __restrict__ ubuf,
    const int* __restrict__ erow, const int* __restrict__ ecol,
    const float* We1, const float* be1, const float* We2, const float* be2,
    const float* Wn1, const float* bn1, const float* Wn2, const float* bn2,
    float* agg) {
  constexpr int K1 = 2 * FX + FE + FU;
  constexpr int KA = pad4(K1);
  constexpr int KN = pad4(FX + EO);
  constexpr int KHE = pad4(EH);
  constexpr int KHN = pad4(NH);
  constexpr int KH = KHE > KHN ? KHE : KHN;
  constexpr int NTN = (NH + 15) / 16, NTNO = (NO + 15) / 16;
  constexpr int NW1 = 16, NW2 = 16, NWN1 = 16 * NTN, NWN2 = 16 * NTNO;

  __shared__ __align__(16) float sA[4][16 * KA];
  __shared__ __align__(16) float sN[4][16 * KN];
  __shared__ __align__(16) float sH[4][16 * KH];
  __shared__ __align__(16) float wE1[KA * NW1], wE2[KHE * NW2];
  __shared__ __align__(16) float wN1[KN * NWN1], wN2[KHN * NWN2];
  __shared__ float bE1[NW1], bE2[NW2], bN1[NWN1], bN2[NWN2];
  __shared__ int sR[4][16], sC[4][16];

  const int w = threadIdx.x >> 5, lane = threadIdx.x & 31;
  const long ebase = ((long)blockIdx.x * 4 + w) * 16;

  stage_w<KA, NW1>(We1, K1, EH, be1, wE1, bE1);
  stage_w<KHE, NW2>(We2, EH, EO, be2, wE2, bE2);
  stage_w<KN, NWN1>(Wn1, FX + EO, NH, bn1, wN1, bN1);
  stage_w<KHN, NWN2>(Wn2, NH, NO, bn2, wN2, bN2);

  if (lane < 16) {
    sR[w][lane] = erow[ebase + lane];
    sC[w][lane] = ecol[ebase + lane];
  }
  __syncthreads();

  // Stage e_in = [x[row] | x[col] | edge_attr | u[batch[row]]], zero-padded.
  for (int t = lane; t < 16 * KA; t += 32) {
    const int m = t / KA, f = t - m * KA;
    const int r = sR[w][m];
    float v = 0.0f;
    if (f < FX)                 v = xin[(long)r * XLD + f];
    else if (f < 2 * FX)        v = xin[(long)sC[w][m] * XLD + (f - FX)];
    else if (f < 2 * FX + FE)   v = ein[(ebase + m) * FE + (f - 2 * FX)];
    else if (f < K1)            v = ubuf[(r / CC) * US + (f - 2 * FX - FE)];
    sA[w][m * KA + f] = v;
  }
  // node1 input = [x[row] | edge_out]; copy x[row] part now.
  for (int t = lane; t < 16 * FX; t += 32) {
    const int m = t / FX, f = t - m * FX;
    sN[w][m * KN + f] = xin[(long)sR[w][m] * XLD + f];
  }
  __syncthreads();

  wave_mlp_gemm<KA, EH, 1, true, KHE>(sA[w], KA, wE1, bE1, sH[w], KH);
  __syncthreads();
  wave_mlp_gemm<KHE, EO, 1, false, KN - FX>(sH[w], KH, wE2, bE2, sN[w] + FX, KN);
  __syncthreads();
  wave_mlp_gemm<KN, NH, NTN, true, KHN>(sN[w], KN, wN1, bN1, sH[w], KH);
  __syncthreads();
  wave_mlp_gemm<KHN, NO, NTNO, false, NO>(sH[w], KH, wN2, bN2, sA[w], KA);
  __syncthreads();

  if (STORE_E) {
    for (int t = lane; t < 16 * EO; t += 32) {
      const int m = t / EO, c = t - m * EO;
      eout[(ebase + m) * ES + c] = sN[w][m * KN + FX + c];
    }
  }
  // scatter-add h into agg[col], edge_attr into edge_u slot, bump counts
  for (int t = lane; t < 16 * NO; t += 32) {
    const int m = t / NO, c = t - m * NO;
    atomicAdd(&agg[(long)sC[w][m] * AGS + c], sA[w][m * KA + c]);
  }
  if (STORE_E) {
    for (int t = lane; t < 16 * EO; t += 32) {
      const int m = t / EO, c = t - m * EO;
      atomicAdd(&agg[(long)sC[w][m] * AGS + 32 + c], sN[w][m * KN + FX + c]);
    }
  }
  if (lane < 16) atomicAdd(&agg[(long)sC[w][lane] * AGS + 44], 1.0f);
}

// ---------------------------------------------------------------------------
// Fused scatter-mean + node2-MLP (+ per-batch partial sums for global MLP).
// ---------------------------------------------------------------------------
template <int FX, int XLD, int NA, int EUA, int FU, int H, int OUT, int OLD,
          bool GSUM, bool APPEND>
__global__ __launch_bounds__(128) void node2_kernel(
    const float* __restrict__ xin, const float* __restrict__ agg,
    const float* __restrict__ ubuf, float* __restrict__ xout, float* gsx,
    float* gseu, const float* __restrict__ deep, const float* __restrict__ x0,
    const float* W1, const float* b1, const float* W2, const float* b2) {
  constexpr int K = FX + NA + FU;
  constexpr int KA = pad4(K);
  constexpr int KH = pad4(H);
  constexpr int NT1 = (H + 15) / 16, NT2 = (OUT + 15) / 16;
  constexpr int NW1 = 16 * NT1, NW2 = 16 * NT2;

  __shared__ __align__(16) float sA[4][16 * KA];
  __shared__ __align__(16) float sH[4][16 * KH];
  __shared__ __align__(16) float sO[4][16 * 32];
  __shared__ __align__(16) float sW1[KA * NW1], sW2[KH * NW2];
  __shared__ float sB1[NW1], sB2[NW2];
  __shared__ float sCnt[4][16];

  const int w = threadIdx.x >> 5, lane = threadIdx.x & 31;
  const long nbase = ((long)blockIdx.x * 4 + w) * 16;

  stage_w<KA, NW1>(W1, K, H, b1, sW1, sB1);
  stage_w<KH, NW2>(W2, H, OUT, b2, sW2, sB2);

  if (lane < 16) sCnt[w][lane] = fmaxf(agg[(nbase + lane) * AGS + 44], 1.0f);
  __syncthreads();

  for (int t = lane; t < 16 * KA; t += 32) {
    const int m = t / KA, f = t - m * KA;
    const long n = nbase + m;
    float v = 0.0f;
    if (f < FX)           v = xin[n * XLD + f];
    else if (f < FX + NA) v = agg[n * AGS + (f - FX)] / sCnt[w][m];
    else if (f < K)       v = ubuf[(n / CC) * US + (f - FX - NA)];
    sA[w][m * KA + f] = v;
  }
  __syncthreads();
  wave_mlp_gemm<KA, H, NT1, true, KH>(sA[w], KA, sW1, sB1, sH[w], KH);
  __syncthreads();
  wave_mlp_gemm<KH, OUT, NT2, false, OUT>(sH[w], KH, sW2, sB2, sO[w], 32);
  __syncthreads();

  for (int t = lane; t < 16 * OUT; t += 32) {
    const int m = t / OUT, c = t - m * OUT;
    const long n = nbase + m;
    const float v = sO[w][m * 32 + c];
    xout[n * OLD + c] = v;
    if (GSUM) atomicAdd(&gsx[(n / CC) * US + c], v);
  }
  if (GSUM) {
    for (int t = lane; t < 16 * EUA; t += 32) {
      const int m = t / EUA, c = t - m * EUA;
      const long n = nbase + m;
      atomicAdd(&gseu[(n / CC) * 16 + c], agg[n * AGS + 32 + c] / sCnt[w][m]);
    }
  }
  if (APPEND && lane < 16) {  // x1 = concat[x2, node_dpg, x_orig[:,10:12]]
    const long n = nbase + lane;
    xout[n * OLD + 20] = deep[n];
    xout[n * OLD + 21] = x0[n * 12 + 10];
    xout[n * OLD + 22] = x0[n * 12 + 11];
  }
}

// ---------------------------------------------------------------------------
// Global MLP: 32 batches = 2 waves of 16 rows, same WMMA chain.
// ---------------------------------------------------------------------------
template <int FU, int NX, int EU, int WK, int H, int GO>
__global__ __launch_bounds__(64) void glob_kernel(
    const float* __restrict__ uin, const float* __restrict__ gsx,
    const float* __restrict__ gseu, float* __restrict__ uout, const float* W1,
    const float* b1, const float* W2, const float* b2) {
  constexpr int K = FU + NX + EU;
  constexpr int KA = pad4(K);
  constexpr int KH = pad4(H);
  constexpr int NT1 = (H + 15) / 16, NT2 = (GO + 15) / 16;
  constexpr int NW1 = 16 * NT1, NW2 = 16 * NT2;

  __shared__ __align__(16) float sA[2][16 * KA];
  __shared__ __align__(16) float sH[2][16 * KH];
  __shared__ __align__(16) float sO[2][16 * 32];
  __shared__ __align__(16) float sW1[KA * NW1], sW2[KH * NW2];
  __shared__ float sB1[NW1], sB2[NW2];

  const int w = threadIdx.x >> 5, lane = threadIdx.x & 31;
  const int bb = w * 16;
  const float inv = 1.0f / (float)CC;

  // WK may exceed K (ml1_glob fi=70 vs 50-dim concat): extra rows hit zeros.
  stage_w<KA, NW1>(W1, WK, H, b1, sW1, sB1);
  stage_w<KH, NW2>(W2, H, GO, b2, sW2, sB2);

  for (int t = lane; t < 16 * KA; t += 32) {
    const int m = t / KA, f = t - m * KA;
    const int b = bb + m;
    float v = 0.0f;
    if (f < FU)           v = uin[b * US + f];
    else if (f < FU + NX) v = gsx[b * US + (f - FU)] * inv;
    else if (f < K)       v = gseu[b * 16 + (f - FU - NX)] * inv;
    sA[w][m * KA + f] = v;
  }
  __syncthreads();
  wave_mlp_gemm<KA, H, NT1, true, KH>(sA[w], KA, sW1, sB1, sH[w], KH);
  __syncthreads();
  wave_mlp_gemm<KH, GO, NT2, false, GO>(sH[w], KH, sW2, sB2, sO[w], 32);
  __syncthreads();
  for (int t = lane; t < 16 * GO; t += 32) {
    const int m = t / GO, c = t - m * GO;
    uout[(bb + m) * US + c] = sO[w][m * 32 + c];
  }
}

__global__ void zero_kernel(float* p, long n) {
  long i = (long)blockIdx.x * blockDim.x + threadIdx.x;
  const long st = (long)gridDim.x * blockDim.x;
  for (; i < n; i += st) p[i] = 0.0f;
}

__global__ void uinit_kernel(const float* __restrict__ uin, float* ub) {
  const int i = threadIdx.x;
  if (i < 32 * US) {
    const int b = i / US, c = i - b * US;
    ub[i] = (c < 20) ? uin[b * 24 + 1 + c] : 0.0f;  // uu = u[:,0,1:21]
  }
}

extern "C" void kernel_launch(void* const* d_in, const int* in_sizes, int n_in,
                              void* d_out, int out_size, void* d_ws,
                              size_t ws_size, hipStream_t stream) {
  // setup_inputs() order: deepgo, x, u, edge, edge_attr, batch, params{12 MLPs x (W1,b1,W2,b2)}
  const float* deep = (const float*)d_in[0];
  const float* x0 = (const float*)d_in[1];
  const float* u_in = (const float*)d_in[2];
  const int* edge = (const int*)d_in[3];
  const float* eattr0 = (const float*)d_in[4];
  auto P = [&](int mlp, int j) { return (const float*)d_in[6 + mlp * 4 + j]; };
  float* out = (float*)d_out;

  const int* erow = edge;
  const int* ecol = edge + EE;

  float* ws = (float*)d_ws;
  float* ebuf = ws;                   // E*10
  float* xbuf = ebuf + EE * ES;       // N*24
  float* aggbuf = xbuf + NN * XS;     // N*48
  float* gsx = aggbuf + NN * AGS;     // 32*32
  float* gseu = gsx + 32 * US;        // 32*16
  float* ubufA = gseu + 32 * 16;      // 32*32
  float* ubufB = ubufA + 32 * US;     // 32*32

  const long zlen = NN * AGS + 32 * US + 32 * 16;
  const dim3 eb((unsigned)(EE / 64)), nb((unsigned)(NN / 64)), thr(128);

  uinit_kernel<<<1, 1024, 0, stream>>>(u_in, ubufA);

  // ---- Meta layer 1: edge 45->5->10, node1 22->15->20, node2 52->20->20, glob ->25->30
  zero_kernel<<<2048, 256, 0, stream>>>(aggbuf, zlen);
  edge_node1_kernel<12, 12, 1, 20, 5, 10, 15, 20, true><<<eb, thr, 0, stream>>>(
      x0, eattr0, ebuf, ubufA, erow, ecol, P(0, 0), P(0, 1), P(0, 2), P(0, 3),
      P(1, 0), P(1, 1), P(1, 2), P(1, 3), aggbuf);
  node2_kernel<12, 12, 20, 10, 20, 20, 20, XS, true, false><<<nb, thr, 0, stream>>>(
      x0, aggbuf, ubufA, xbuf, gsx, gseu, deep, x0, P(2, 0), P(2, 1), P(2, 2), P(2, 3));
  glob_kernel<20, 20, 10, 70, 25, 30><<<1, 64, 0, stream>>>(
      ubufA, gsx, gseu, ubufB, P(3, 0), P(3, 1), P(3, 2), P(3, 3));

  // ---- Meta layer 2: edge 80->5->10, node1 30->20->25, node2 75->25->20, glob 60->25->20
  zero_kernel<<<2048, 256, 0, stream>>>(aggbuf, zlen);
  edge_node1_kernel<20, XS, 10, 30, 5, 10, 20, 25, true><<<eb, thr, 0, stream>>>(
      xbuf, ebuf, ebuf, ubufB, erow, ecol, P(4, 0), P(4, 1), P(4, 2), P(4, 3),
      P(5, 0), P(5, 1), P(5, 2), P(5, 3), aggbuf);
  node2_kernel<20, XS, 25, 10, 30, 25, 20, XS, true, true><<<nb, thr, 0, stream>>>(
      xbuf, aggbuf, ubufB, xbuf, gsx, gseu, deep, x0, P(6, 0), P(6, 1), P(6, 2), P(6, 3));
  glob_kernel<30, 20, 10, 60, 25, 20><<<1, 64, 0, stream>>>(
      ubufB, gsx, gseu, ubufA, P(7, 0), P(7, 1), P(7, 2), P(7, 3));

  // ---- Meta layer 3: x=23 feats, edge 76->5->10, node1 33->23->10, node2 53->10->1
  // ml5_glob output is discarded by the reference -> skipped entirely.
  zero_kernel<<<2048, 256, 0, stream>>>(aggbuf, zlen);
  edge_node1_kernel<23, XS, 10, 20, 5, 10, 23, 10, false><<<eb, thr, 0, stream>>>(
      xbuf, ebuf, ebuf, ubufA, erow, ecol, P(8, 0), P(8, 1), P(8, 2), P(8, 3),
      P(9, 0), P(9, 1), P(9, 2), P(9, 3), aggbuf);
  node2_kernel<23, XS, 10, 10, 20, 10, 1, 1, false, false><<<nb, thr, 0, stream>>>(
      xbuf, aggbuf, ubufA, out, gsx, gseu, deep, x0, P(10, 0), P(10, 1), P(10, 2), P(10, 3));

  (void)in_sizes; (void)n_in; (void)out_size; (void)ws_size;
}